// CHSLoss_34127810134739
// MI455X (gfx1250) — compile-verified
//
#include <hip/hip_runtime.h>
#include <hip/hip_bf16.h>

typedef __attribute__((ext_vector_type(2))) float v2f;
typedef __attribute__((ext_vector_type(8))) float v8f;

#define B_   32
#define HW_  16384          // 128*128 per batch
#define TOT_ (B_ * HW_)     // 524288
#define GTW_ 1024           // gt row pitch (floats)
#define NBLK_MSE 256
#define NTHR 256

// ---------------------------------------------------------------------------
// Stage 1: 8x8 block-sum pooling of gt_density [32,1024,1024] -> dg [32,16384]
// Pure streaming: 134 MB read, float4 loads, prefetch ahead.
// ---------------------------------------------------------------------------
__global__ void pool_kernel(const float* __restrict__ gt, float* __restrict__ dg_all) {
    int idx = blockIdx.x * NTHR + threadIdx.x;          // 0 .. 524287
    int b   = idx >> 14;
    int pos = idx & (HW_ - 1);
    int r   = pos >> 7;
    int c   = pos & 127;
    const float* base = gt + (size_t)b * (GTW_ * GTW_) + (size_t)(r * 8) * GTW_ + c * 8;
    float acc = 0.0f;
#pragma unroll
    for (int rr = 0; rr < 8; ++rr) {
        const float4* p = (const float4*)(base + rr * GTW_);
        float4 x = p[0];
        float4 y = p[1];
        __builtin_prefetch(base + rr * GTW_ + 8 * GTW_, 0, 1); // next 8-row block
        acc += (x.x + x.y) + (x.z + x.w) + (y.x + y.y) + (y.z + y.w);
    }
    dg_all[idx] = acc;
}

// ---------------------------------------------------------------------------
// Stage 2: exact k-th largest of err = |dg - d| per (batch, map) via 4-pass
// radix select on float bit patterns (all values >= 0 so uint order == float
// order). One 256-thread workgroup per (b, m). Writes thr[m*32+b].
// num < 1  =>  thr = +inf (where() then always picks dg == plain MSE branch).
// ---------------------------------------------------------------------------
__global__ void thresh_kernel(const float* __restrict__ dconv,
                              const float* __restrict__ dtran,
                              const float* __restrict__ dg_all,
                              const int*   __restrict__ process,
                              float* __restrict__ thr) {
    int b = blockIdx.x;      // 0..31
    int m = blockIdx.y;      // 0: conv, 1: tran
    const float* d = (m == 0 ? dconv : dtran) + (size_t)b * HW_;
    const float* g = dg_all + (size_t)b * HW_;

    int p   = process[0];
    int num = (int)((double)HW_ * 0.1 * (double)p);   // matches Python int() trunc

    if (num < 1) {
        if (threadIdx.x == 0) thr[m * B_ + b] = __builtin_inff();
        return;
    }

    __shared__ unsigned hist[256];
    __shared__ unsigned s_sel;
    __shared__ unsigned s_rem;

    unsigned prefix    = 0u;
    unsigned remaining = (unsigned)num;

    for (int pass = 0; pass < 4; ++pass) {
        int shift = 24 - 8 * pass;
        hist[threadIdx.x] = 0u;
        __syncthreads();

        unsigned mask = (pass == 0) ? 0u : (0xFFFFFFFFu << (shift + 8));
        for (int i = threadIdx.x; i < HW_; i += NTHR) {
            unsigned u = __float_as_uint(fabsf(g[i] - d[i]));
            if ((u & mask) == prefix)
                atomicAdd(&hist[(u >> shift) & 255u], 1u);
        }
        __syncthreads();

        if (threadIdx.x == 0) {
            unsigned cum = 0u, sel = 0u, rem = 1u;
            for (int dd = 255; dd >= 0; --dd) {
                cum += hist[dd];
                if (cum >= remaining) {
                    sel = (unsigned)dd;
                    rem = remaining - (cum - hist[dd]);
                    break;
                }
            }
            s_sel = sel;
            s_rem = rem;
        }
        __syncthreads();
        prefix   |= (s_sel << shift);
        remaining = s_rem;
        __syncthreads();
    }

    if (threadIdx.x == 0) thr[m * B_ + b] = __uint_as_float(prefix);
}

// ---------------------------------------------------------------------------
// Stage 3: masked-MSE partial sums. 256 blocks x 256 thr, 8 elements each.
// Deterministic: fixed per-thread order, fixed shuffle/LDS tree.
// ---------------------------------------------------------------------------
__global__ void mse_partial_kernel(const float* __restrict__ dconv,
                                   const float* __restrict__ dtran,
                                   const float* __restrict__ dg_all,
                                   const float* __restrict__ thr,
                                   const int*   __restrict__ process,
                                   float* __restrict__ partial) {
    float w  = 1.0f * (float)process[0];     // MAX_WEIGHT_RATIO * process
    float iw = 1.0f - w;

    int tid    = blockIdx.x * NTHR + threadIdx.x;
    int stride = NBLK_MSE * NTHR;            // 65536

    float acc = 0.0f;
    for (int i = tid; i < TOT_; i += stride) {
        int   b     = i >> 14;
        float gd    = dg_all[i];
        float dc    = dconv[i];
        float dt    = dtran[i];
        float err_c = fabsf(gd - dc);
        float err_t = fabsf(gd - dt);
        float cmb_c = w * dc + iw * gd;
        float cmb_t = w * dt + iw * gd;
        float sup_ft = (err_c >= thr[b])      ? cmb_t : gd;  // supervise conv from tran
        float sup_fc = (err_t >= thr[B_ + b]) ? cmb_c : gd;  // supervise tran from conv
        float e1 = dc - sup_ft;
        float e2 = dt - sup_fc;
        acc += e1 * e1 + e2 * e2;
    }

    // wave32 butterfly, then cross-wave via LDS, fixed order
    for (int off = 16; off >= 1; off >>= 1) acc += __shfl_xor(acc, off, 32);
    __shared__ float wsum[NTHR / 32];
    int lane = threadIdx.x & 31;
    int wav  = threadIdx.x >> 5;
    if (lane == 0) wsum[wav] = acc;
    __syncthreads();
    if (threadIdx.x == 0) {
        float s = 0.0f;
        for (int k = 0; k < NTHR / 32; ++k) s += wsum[k];
        partial[blockIdx.x] = s;
    }
}

// ---------------------------------------------------------------------------
// Stage 4: single-wave final reduce of 256 partials via chained
// V_WMMA_F32_16X16X4_F32 with ones A-matrix. D rows 0 and 8 each hold the
// exact column sums (bitwise identical accumulation), so a full 32-lane
// butterfly yields 2x the total; *0.5f is exact. A=ones makes the result
// independent of the B-operand lane layout.
// ---------------------------------------------------------------------------
__global__ void final_reduce_kernel(const float* __restrict__ partial,
                                    float* __restrict__ out) {
    int L = threadIdx.x;              // 0..31, EXEC all ones, no divergence
    v2f a; a[0] = 1.0f; a[1] = 1.0f;
    v8f c = {};
#pragma unroll
    for (int chunk = 0; chunk < 4; ++chunk) {
        v2f bv;
        bv[0] = partial[chunk * 64 + L];
        bv[1] = partial[chunk * 64 + 32 + L];
        c = __builtin_amdgcn_wmma_f32_16x16x4_f32(
                /*neg_a=*/false, a, /*neg_b=*/false, bv,
                /*c_mod=*/(short)0, c, /*reuse_a=*/false, /*reuse_b=*/false);
    }
    float s = c[0];
    for (int off = 16; off >= 1; off >>= 1) s += __shfl_xor(s, off, 32);
    if (L == 0) out[0] = 0.5f * s;
}

// ---------------------------------------------------------------------------
extern "C" void kernel_launch(void* const* d_in, const int* in_sizes, int n_in,
                              void* d_out, int out_size, void* d_ws, size_t ws_size,
                              hipStream_t stream) {
    const float* dconv   = (const float*)d_in[0];   // [32,1,128,128]
    const float* dtran   = (const float*)d_in[1];   // [32,1,128,128]
    const float* gt      = (const float*)d_in[2];   // [32,1,1024,1024]
    const int*   process = (const int*)  d_in[3];   // scalar

    float* dg_all  = (float*)d_ws;                  // 524288 floats (2 MB)
    float* thr     = dg_all + TOT_;                 // 64 floats [m][b]
    float* partial = thr + 2 * B_;                  // 256 floats
    float* out     = (float*)d_out;                 // scalar fp32

    // Stage 1: pooling (bandwidth-dominant, ~134 MB streamed)
    pool_kernel<<<TOT_ / NTHR, NTHR, 0, stream>>>(gt, dg_all);

    // Stage 2: exact per-row k-th-largest thresholds (L2-resident)
    dim3 g2(B_, 2);
    thresh_kernel<<<g2, NTHR, 0, stream>>>(dconv, dtran, dg_all, process, thr);

    // Stage 3: masked MSE partials (deterministic)
    mse_partial_kernel<<<NBLK_MSE, NTHR, 0, stream>>>(dconv, dtran, dg_all, thr,
                                                      process, partial);

    // Stage 4: WMMA-based single-wave final reduction
    final_reduce_kernel<<<1, 32, 0, stream>>>(partial, out);
}